// GCN_58506044506615
// MI455X (gfx1250) — compile-verified
//
#include <hip/hip_runtime.h>

// GCN 2-layer forward for MI455X (gfx1250, wave32).
// GEMMs use V_WMMA_F32_16X16X4_F32 (full fp32 precision; GEMM FLOPs are
// negligible so the fp32 matrix rate is more than enough). Scatters use
// native f32 global atomics; accumulators are L2-resident (192MB L2).

typedef float v2f __attribute__((ext_vector_type(2)));
typedef float v8f __attribute__((ext_vector_type(8)));

#define NFEAT 128
#define NHID  64
#define NCLASS 40

// ---------------- elementwise helpers ----------------

__global__ void fill_kernel(float* __restrict__ p, float v, int n) {
  int i = blockIdx.x * blockDim.x + threadIdx.x;
  if (i < n) p[i] = v;
}

__global__ void deg_scatter_kernel(const long long* __restrict__ col,
                                   const float* __restrict__ w,
                                   float* __restrict__ deg, int E) {
  int e = blockIdx.x * blockDim.x + threadIdx.x;
  if (e < E) unsafeAtomicAdd(&deg[(int)col[e]], w[e]);
}

__global__ void dinv_kernel(const float* __restrict__ deg,
                            float* __restrict__ dinv, int n) {
  int i = blockIdx.x * blockDim.x + threadIdx.x;
  if (i < n) {
    float d = deg[i];
    dinv[i] = (d > 0.0f) ? rsqrtf(d) : 0.0f;
  }
}

// out[i][f] = dinv[i]^2 * h[i][f]   (self-loop term, also initializes out)
template <int F>
__global__ void self_init_kernel(const float* __restrict__ h,
                                 const float* __restrict__ dinv,
                                 float* __restrict__ out, int total) {
  int t = blockIdx.x * blockDim.x + threadIdx.x;
  if (t >= total) return;
  int i = t / F;
  float di = dinv[i];
  out[t] = di * di * h[t];
}

// out[i][f] = act(in[i][f] + b[f])
template <int F, bool RELU>
__global__ void bias_act_kernel(const float* __restrict__ in,
                                const float* __restrict__ b,
                                float* __restrict__ out, int total) {
  int t = blockIdx.x * blockDim.x + threadIdx.x;
  if (t >= total) return;
  float v = in[t] + b[t % F];
  out[t] = RELU ? fmaxf(v, 0.0f) : v;
}

// ---------------- edge scatter: out[col] += dinv[r]*w*dinv[c] * h[row] ----------------
// F/4 threads per edge; each thread does one float4 gather + 4 native f32 atomics.
// 16 lanes (F=64) gather 256B contiguous -> fully coalesced h-row reads.
template <int F>
__global__ void edge_scatter_kernel(const long long* __restrict__ row,
                                    const long long* __restrict__ col,
                                    const float* __restrict__ w,
                                    const float* __restrict__ dinv,
                                    const float* __restrict__ h,
                                    float* __restrict__ out, int E) {
  constexpr int TPE = F / 4;
  long long t = (long long)blockIdx.x * blockDim.x + threadIdx.x;
  int e = (int)(t / TPE);
  int part = (int)(t % TPE);
  if (e >= E) return;
  int r = (int)row[e];
  int c = (int)col[e];
  float norm = dinv[r] * w[e] * dinv[c];
  const float4 hv = *(const float4*)(h + (size_t)r * F + part * 4);
  float* op = out + (size_t)c * F + part * 4;
  unsafeAtomicAdd(op + 0, norm * hv.x);
  unsafeAtomicAdd(op + 1, norm * hv.y);
  unsafeAtomicAdd(op + 2, norm * hv.z);
  unsafeAtomicAdd(op + 3, norm * hv.w);
}

// ---------------- fp32 WMMA GEMM: C[M x NREAL] = A[M x K] * B[K x NREAL] ----------------
// One 16x16 output tile per wave, K stepped by 4 (V_WMMA_F32_16X16X4_F32).
// A-fragment (16x4 f32): lanes 0-15 hold K={k,k+1}, lanes 16-31 hold K={k+2,k+3}.
// B-fragment (4x16 f32): symmetric, lane&15 selects column N.
// C/D (16x16 f32, 8 VGPRs): c[i] is row (mt*16 + i + (lane>>4)*8), col (nt*16 + lane&15).
// M must be a multiple of 16 (100000 = 6250*16). NPAD pads NREAL up to 16.
template <int K, int NPAD, int NREAL>
__global__ void gemm_wmma_kernel(const float* __restrict__ A,
                                 const float* __restrict__ B,
                                 float* __restrict__ C, int mtiles) {
  constexpr int NT = NPAD / 16;
  int wave = threadIdx.x >> 5;
  int lane = threadIdx.x & 31;
  int tile = blockIdx.x * (blockDim.x >> 5) + wave;
  if (tile >= mtiles * NT) return;  // wave-uniform: EXEC stays all-ones for WMMA
  int mt = tile / NT;
  int nt = tile - mt * NT;
  int m = mt * 16 + (lane & 15);
  int n = nt * 16 + (lane & 15);
  int kh = (lane >> 4) * 2;  // 0 for lanes 0-15, 2 for lanes 16-31
  const bool nok = (NREAL == NPAD) || (n < NREAL);

  v8f c = {0.f, 0.f, 0.f, 0.f, 0.f, 0.f, 0.f, 0.f};
  const float* arow = A + (size_t)m * K;
#pragma unroll
  for (int k = 0; k < K; k += 4) {
    v2f a, b;
    a[0] = arow[k + kh];
    a[1] = arow[k + kh + 1];
    b[0] = nok ? B[(size_t)(k + kh) * NREAL + n] : 0.0f;
    b[1] = nok ? B[(size_t)(k + kh + 1) * NREAL + n] : 0.0f;
    c = __builtin_amdgcn_wmma_f32_16x16x4_f32(
        /*neg_a=*/false, a, /*neg_b=*/false, b,
        /*c_mod=*/(short)0, c, /*reuse_a=*/false, /*reuse_b=*/false);
  }
  if (nok) {
    int rb = mt * 16 + (lane >> 4) * 8;
#pragma unroll
    for (int i = 0; i < 8; ++i)
      C[(size_t)(rb + i) * NREAL + n] = c[i];
  }
}

// ---------------- launch ----------------

static inline unsigned cdiv(long long a, long long b) { return (unsigned)((a + b - 1) / b); }

extern "C" void kernel_launch(void* const* d_in, const int* in_sizes, int n_in,
                              void* d_out, int out_size, void* d_ws, size_t ws_size,
                              hipStream_t stream) {
  const float*      x   = (const float*)d_in[0];
  const long long*  ei  = (const long long*)d_in[1];   // int64 per reference
  const float*      ew  = (const float*)d_in[2];
  const float*      W1  = (const float*)d_in[3];
  const float*      b1  = (const float*)d_in[4];
  const float*      W2  = (const float*)d_in[5];
  const float*      b2  = (const float*)d_in[6];
  float*            out = (float*)d_out;

  const int N = in_sizes[0] / NFEAT;   // 100000
  const int E = in_sizes[1] / 2;       // 1600000
  const long long* rowp = ei;          // edge_index[0] = sources
  const long long* colp = ei + E;      // edge_index[1] = targets

  // Carve workspace (floats), 256B-aligned chunks. Total ~68 MB.
  float* ws = (float*)d_ws;
  size_t off = 0;
  auto alloc = [&](size_t nelem) {
    float* p = ws + off;
    off += (nelem + 63) & ~(size_t)63;
    return p;
  };
  float* deg  = alloc((size_t)N);
  float* dinv = alloc((size_t)N);
  float* h1   = alloc((size_t)N * NHID);    // x@W1, later reused for relu(out1+b1)
  float* o1   = alloc((size_t)N * NHID);    // layer-1 aggregation
  float* h2   = alloc((size_t)N * NCLASS);  // h@W2

  const int TB = 256;

  // 1) degree = 1 (self loop) + scatter of edge weights over col
  fill_kernel<<<cdiv(N, TB), TB, 0, stream>>>(deg, 1.0f, N);
  deg_scatter_kernel<<<cdiv(E, TB), TB, 0, stream>>>(colp, ew, deg, E);
  dinv_kernel<<<cdiv(N, TB), TB, 0, stream>>>(deg, dinv, N);

  const int mtiles = N / 16;  // 6250, exact

  // 2) layer 1: h1 = x @ W1  (K=128, N=64)
  {
    unsigned tiles = (unsigned)mtiles * (NHID / 16);
    gemm_wmma_kernel<NFEAT, NHID, NHID>
        <<<cdiv(tiles, TB / 32), TB, 0, stream>>>(x, W1, h1, mtiles);
  }
  // o1 = dinv^2 * h1  (self-loop term, initializes accumulator)
  self_init_kernel<NHID><<<cdiv((long long)N * NHID, TB), TB, 0, stream>>>(
      h1, dinv, o1, N * NHID);
  // o1[col] += dinv[r]*w*dinv[c] * h1[row]
  edge_scatter_kernel<NHID>
      <<<cdiv((long long)E * (NHID / 4), TB), TB, 0, stream>>>(
          rowp, colp, ew, dinv, h1, o1, E);
  // h1 = relu(o1 + b1)   (reuse h1 buffer as layer-2 input)
  bias_act_kernel<NHID, true><<<cdiv((long long)N * NHID, TB), TB, 0, stream>>>(
      o1, b1, h1, N * NHID);

  // 3) layer 2: h2 = h1 @ W2  (K=64, N=40 padded to 48)
  {
    unsigned tiles = (unsigned)mtiles * (48 / 16);
    gemm_wmma_kernel<NHID, 48, NCLASS>
        <<<cdiv(tiles, TB / 32), TB, 0, stream>>>(h1, W2, h2, mtiles);
  }
  // out = dinv^2 * h2
  self_init_kernel<NCLASS><<<cdiv((long long)N * NCLASS, TB), TB, 0, stream>>>(
      h2, dinv, out, N * NCLASS);
  // out[col] += norm * h2[row]
  edge_scatter_kernel<NCLASS>
      <<<cdiv((long long)E * (NCLASS / 4), TB), TB, 0, stream>>>(
          rowp, colp, ew, dinv, h2, out, E);
  // out += b2
  bias_act_kernel<NCLASS, false><<<cdiv((long long)N * NCLASS, TB), TB, 0, stream>>>(
      out, b2, out, N * NCLASS);
}